// AttentionHeteroRGCNLayer_86715389706549
// MI455X (gfx1250) — compile-verified
//
#include <hip/hip_runtime.h>
#include <hip/hip_bf16.h>

// ---------------------------------------------------------------------------
// AttentionHeteroRGCNLayer for MI455X (gfx1250, wave32, WMMA)
//
//  * softmax over a 1-element tensor == 1.0 exactly; softmax([1,1,1]) ==
//    [1/3,1/3,1/3] exactly -> relation weights are constant 1/3 regardless of
//    a0/a1/a2.   h = feat @ ((W0+W1+W2)/3)
//  * GEMM: V_WMMA_F32_16X16X4_F32, 2 row-tiles x 8 col-tiles per wave,
//    B-matrix pre-interleaved so each B fragment is one b64 load.
//  * Aggregation: device-built CSR (count + scan + fill), then a gather
//    kernel that accumulates in registers (no float atomics; h stays in L2)
//    fused with ReLU + LayerNorm.
// ---------------------------------------------------------------------------

#define N_NODES   50000
#define DIM       256
#define ROW_TILES (N_NODES / 16)          // 3125
#define NB        ((N_NODES + 255) / 256) // 196 scan blocks per relation
#define LN_EPS    1e-5f

typedef __attribute__((ext_vector_type(2))) float v2f;
typedef __attribute__((ext_vector_type(8))) float v8f;

// ---------------------------------------------------------------------------
__global__ void k_zero_i32(int* __restrict__ p, int n) {
    int i = blockIdx.x * blockDim.x + threadIdx.x;
    if (i < n) p[i] = 0;
}

// ---------------------------------------------------------------------------
// Wi: folded weights, row-pair interleaved:
//   Wi[((k>>1)*DIM + n)*2 + (k&1)] = (W0[k,n]+W1[k,n]+W2[k,n])/3
// so a B fragment {W[k][n], W[k+1][n]} is one contiguous 8-byte load.
// ---------------------------------------------------------------------------
__global__ void k_prep_w(const float* __restrict__ W0,
                         const float* __restrict__ W1,
                         const float* __restrict__ W2,
                         float* __restrict__ Wi) {
    int i = blockIdx.x * blockDim.x + threadIdx.x;
    if (i < DIM * DIM) {
        int k = i >> 8, n = i & 255;
        float v = (W0[i] + W1[i] + W2[i]) * (1.0f / 3.0f);
        Wi[(((size_t)(k >> 1) * DIM + n) << 1) + (k & 1)] = v;
    }
}

// ---------------------------------------------------------------------------
// H[N,256] = feat[N,256] @ Wc[256,256] via V_WMMA_F32_16X16X4_F32.
// Block = 256 threads = 8 waves; block covers 128 rows x 256 cols.
//   wave w: row-tile pair (w>>1), column half (w&1)*128.
// Fragment layouts (ISA 7.12.2, 32-bit):
//   A 16x4 : lane m = lane&15; lanes<16 K={k0,k0+1}, lanes>=16 K={k0+2,k0+3}
//   B 4x16 : lane n = lane&15; same K split -> interleaved b64 from Wi
//   C 16x16: VGPR i -> row (i + 8*(lane>=16)), col lane&15
// ---------------------------------------------------------------------------
__global__ __launch_bounds__(256) void k_gemm_wmma(
        const float* __restrict__ A,
        const float* __restrict__ Wi,
        float* __restrict__ H) {
    const int lane = threadIdx.x & 31;
    const int wave = threadIdx.x >> 5;
    const int rt0  = blockIdx.x * 8 + (wave >> 1) * 2;
    if (rt0 >= ROW_TILES) return;            // wave-uniform exit, EXEC stays full
    const int  rt1 = rt0 + 1;
    const bool v1  = (rt1 < ROW_TILES);
    const int  n0  = (wave & 1) * 128;
    const int  mA  = lane & 15;
    const int  kOff = (lane >> 4) * 2;       // 0 for lanes 0-15, 2 for 16-31
    const int  nB  = lane & 15;

    v8f acc0[8] = {};
    v8f acc1[8] = {};

    const int row0 = rt0 * 16 + mA;
    const int row1 = v1 ? (rt1 * 16 + mA) : row0;  // clamp for safe OOB loads
    const float* A0 = A + (size_t)row0 * DIM;
    const float* A1 = A + (size_t)row1 * DIM;

    for (int k0 = 0; k0 < DIM; k0 += 4) {
        v2f a0, a1;
        a0.x = A0[k0 + kOff];  a0.y = A0[k0 + kOff + 1];
        a1.x = A1[k0 + kOff];  a1.y = A1[k0 + kOff + 1];
        // k' = (k0 + kOff)/2
        const float* Wp = Wi + ((size_t)((k0 >> 1) + (lane >> 4)) * DIM) * 2;
#pragma unroll
        for (int ct = 0; ct < 8; ++ct) {
            const int n = n0 + ct * 16 + nB;
            v2f b = *(const v2f*)(Wp + 2 * n);   // {W[k][n], W[k+1][n]}
            acc0[ct] = __builtin_amdgcn_wmma_f32_16x16x4_f32(
                false, a0, false, b, (short)0, acc0[ct], false, false);
            acc1[ct] = __builtin_amdgcn_wmma_f32_16x16x4_f32(
                false, a1, false, b, (short)0, acc1[ct], false, false);
        }
    }

    const int rb0 = rt0 * 16 + (lane >> 4) * 8;
#pragma unroll
    for (int ct = 0; ct < 8; ++ct) {
        float* Hp = H + (size_t)rb0 * DIM + (n0 + ct * 16 + nB);
#pragma unroll
        for (int i = 0; i < 8; ++i)
            Hp[(size_t)i * DIM] = acc0[ct][i];
    }
    if (v1) {
        const int rb1 = rt1 * 16 + (lane >> 4) * 8;
#pragma unroll
        for (int ct = 0; ct < 8; ++ct) {
            float* Hp = H + (size_t)rb1 * DIM + (n0 + ct * 16 + nB);
#pragma unroll
            for (int i = 0; i < 8; ++i)
                Hp[(size_t)i * DIM] = acc1[ct][i];
        }
    }
}

// ---------------------------------------------------------------------------
// Per-relation in-degree (int atomics)
// ---------------------------------------------------------------------------
__global__ void k_degree(const int* __restrict__ dst, int* __restrict__ deg, int E) {
    int i = blockIdx.x * blockDim.x + threadIdx.x;
    if (i < E) atomicAdd(&deg[dst[i]], 1);
}

// ---------------------------------------------------------------------------
// Block-wide exclusive scan of one int per thread (256 threads, wave32)
// ---------------------------------------------------------------------------
__device__ __forceinline__ int block_scan_excl(int v, int tid) {
    const int lane = tid & 31, wv = tid >> 5;
    int inc = v;
#pragma unroll
    for (int d = 1; d < 32; d <<= 1) {
        int t = __shfl_up(inc, d, 32);
        if (lane >= d) inc += t;
    }
    __shared__ int wsum[8];
    __shared__ int wpre[8];
    if (lane == 31) wsum[wv] = inc;
    __syncthreads();
    if (tid == 0) {
        int s = 0;
#pragma unroll
        for (int j = 0; j < 8; ++j) { wpre[j] = s; s += wsum[j]; }
    }
    __syncthreads();
    return inc - v + wpre[wv];
}

// pass 1: per-block exclusive scan of deg -> off, block totals -> bsum
__global__ __launch_bounds__(256) void k_scan1(const int* __restrict__ deg,
                                               int* __restrict__ off,
                                               int* __restrict__ bsum) {
    const int r = blockIdx.y;
    const int i = blockIdx.x * 256 + threadIdx.x;
    const int v = (i < N_NODES) ? deg[r * N_NODES + i] : 0;
    const int e = block_scan_excl(v, threadIdx.x);
    if (i < N_NODES) off[r * N_NODES + i] = e;
    if (threadIdx.x == 255) bsum[r * 256 + blockIdx.x] = e + v;
}

// pass 2: exclusive scan of the NB block totals (one block per relation)
__global__ __launch_bounds__(256) void k_scan2(int* __restrict__ bsum) {
    const int r = blockIdx.x;
    const int t = threadIdx.x;
    const int v = (t < NB) ? bsum[r * 256 + t] : 0;
    const int e = block_scan_excl(v, t);
    if (t < NB) bsum[r * 256 + t] = e;
}

// pass 3: add block bases -> final offsets; init cursors
__global__ __launch_bounds__(256) void k_scan3(int* __restrict__ off,
                                               const int* __restrict__ bsum,
                                               int* __restrict__ cursor) {
    const int r = blockIdx.y;
    const int i = blockIdx.x * 256 + threadIdx.x;
    if (i < N_NODES) {
        const int v = off[r * N_NODES + i] + bsum[r * 256 + blockIdx.x];
        off[r * N_NODES + i]    = v;
        cursor[r * N_NODES + i] = v;
    }
}

// ---------------------------------------------------------------------------
// CSR fill: one int atomic per edge
// ---------------------------------------------------------------------------
__global__ void k_fill(const int* __restrict__ src, const int* __restrict__ dst,
                       int* __restrict__ cursor, int* __restrict__ csr, int E) {
    int i = blockIdx.x * blockDim.x + threadIdx.x;
    if (i < E) {
        int pos = atomicAdd(&cursor[dst[i]], 1);
        csr[pos] = src[i];
    }
}

// ---------------------------------------------------------------------------
// Gather + cross-relation mean + ReLU + LayerNorm, one block per node.
// h rows are read from L2 (h = 51 MB < 192 MB L2); aggregate lives in regs.
// ---------------------------------------------------------------------------
__global__ __launch_bounds__(256) void k_gather_ln(
        const float* __restrict__ H,
        const int*   __restrict__ deg,
        const int*   __restrict__ off,
        const int*   __restrict__ csr,
        const float* __restrict__ gamma,
        const float* __restrict__ beta,
        float*       __restrict__ out, int E) {
    const int node = blockIdx.x;
    const int tid  = threadIdx.x;

    float acc = 0.0f;
#pragma unroll
    for (int r = 0; r < 3; ++r) {
        const int dg = deg[r * N_NODES + node];
        const int o  = off[r * N_NODES + node];
        const int* lst = csr + (size_t)r * E + o;
        float s = 0.0f;
        for (int e = 0; e < dg; ++e) {
            const int sn = lst[e];                      // uniform (scalarizes)
            s += H[(size_t)sn * DIM + tid];             // coalesced 1KB row
        }
        acc += s * (1.0f / (3.0f * fmaxf((float)dg, 1.0f)));
    }

    const float x = fmaxf(acc, 0.0f);

    // LayerNorm over the 256 dims held one-per-thread
    __shared__ float s_sum[8];
    __shared__ float s_sq[8];
    const int lane = tid & 31, wv = tid >> 5;
    float sum = x, sq = x * x;
#pragma unroll
    for (int o2 = 16; o2 > 0; o2 >>= 1) {
        sum += __shfl_down(sum, o2, 32);
        sq  += __shfl_down(sq,  o2, 32);
    }
    if (lane == 0) { s_sum[wv] = sum; s_sq[wv] = sq; }
    __syncthreads();
    if (wv == 0) {
        float a = (lane < 8) ? s_sum[lane] : 0.0f;
        float b = (lane < 8) ? s_sq[lane]  : 0.0f;
#pragma unroll
        for (int o2 = 4; o2 > 0; o2 >>= 1) {
            a += __shfl_down(a, o2, 32);
            b += __shfl_down(b, o2, 32);
        }
        if (lane == 0) { s_sum[0] = a; s_sq[0] = b; }
    }
    __syncthreads();
    const float mu  = s_sum[0] * (1.0f / DIM);
    const float var = s_sq[0] * (1.0f / DIM) - mu * mu;
    out[(size_t)node * DIM + tid] =
        (x - mu) * rsqrtf(var + LN_EPS) * gamma[tid] + beta[tid];
}

// ---------------------------------------------------------------------------
extern "C" void kernel_launch(void* const* d_in, const int* in_sizes, int n_in,
                              void* d_out, int out_size, void* d_ws, size_t ws_size,
                              hipStream_t stream) {
    const float* feat  = (const float*)d_in[0];
    const float* W0    = (const float*)d_in[1];
    const float* W1    = (const float*)d_in[2];
    const float* W2    = (const float*)d_in[3];
    // d_in[4..6] = a0,a1,a2: softmax of a singleton == 1 exactly, so the
    // relation weights are the constant 1/3 regardless of their values.
    const float* gamma = (const float*)d_in[7];
    const float* beta  = (const float*)d_in[8];
    const int* srcs[3] = {(const int*)d_in[9],  (const int*)d_in[11], (const int*)d_in[13]};
    const int* dsts[3] = {(const int*)d_in[10], (const int*)d_in[12], (const int*)d_in[14]};
    const int  E       = in_sizes[9];

    float* out = (float*)d_out;

    // workspace layout: Wi | H | deg[3N] | off[3N] | cursor[3N] | bsum[3*256] | csr[3E]
    float* Wi     = (float*)d_ws;
    float* H      = Wi + DIM * DIM;
    int*   deg    = (int*)(H + (size_t)N_NODES * DIM);
    int*   off    = deg + 3 * N_NODES;
    int*   cursor = off + 3 * N_NODES;
    int*   bsum   = cursor + 3 * N_NODES;
    int*   csr    = bsum + 3 * 256;

    // 1) zero degree counters
    k_zero_i32<<<(3 * N_NODES + 255) / 256, 256, 0, stream>>>(deg, 3 * N_NODES);

    // 2) fold + interleave the relation weight matrices
    k_prep_w<<<(DIM * DIM + 255) / 256, 256, 0, stream>>>(W0, W1, W2, Wi);

    // 3) H = feat @ Wc via f32 WMMA (2 row-tiles x 128 cols per wave)
    k_gemm_wmma<<<(ROW_TILES + 7) / 8, 256, 0, stream>>>(feat, Wi, H);

    // 4) per-relation degrees
    for (int r = 0; r < 3; ++r)
        k_degree<<<(E + 255) / 256, 256, 0, stream>>>(dsts[r], deg + (size_t)r * N_NODES, E);

    // 5) exclusive scan -> CSR offsets, cursors
    {
        dim3 g1(NB, 3);
        k_scan1<<<g1, 256, 0, stream>>>(deg, off, bsum);
        k_scan2<<<3, 256, 0, stream>>>(bsum);
        k_scan3<<<g1, 256, 0, stream>>>(off, bsum, cursor);
    }

    // 6) CSR fill (1 int atomic per edge)
    for (int r = 0; r < 3; ++r)
        k_fill<<<(E + 255) / 256, 256, 0, stream>>>(
            srcs[r], dsts[r], cursor + (size_t)r * N_NODES, csr + (size_t)r * E, E);

    // 7) gather + mean + ReLU + LayerNorm (no float atomics, single output write)
    k_gather_ln<<<N_NODES, 256, 0, stream>>>(H, deg, off, csr, gamma, beta, out, E);
}